// Attention_6459630813972
// MI455X (gfx1250) — compile-verified
//
#include <hip/hip_runtime.h>
#include <hip/hip_bf16.h>
#include <math.h>

typedef __attribute__((ext_vector_type(2))) float v2f;
typedef __attribute__((ext_vector_type(8))) float v8f;

#define REL_DIM 690
#define NUM_CLASSES 53
#define BLK 256
#define NWAVES 8
#define WCHUNK 512

// ---------------------------------------------------------------------------
// Kernel 1: combined[c,d] = attention_weight[c,d] * relation_weight[c,d]
// ---------------------------------------------------------------------------
__global__ void combine_kernel(const float* __restrict__ aw,
                               const float* __restrict__ rw,
                               float* __restrict__ comb, int n) {
    int i = blockIdx.x * blockDim.x + threadIdx.x;
    if (i < n) comb[i] = aw[i] * rw[i];
}

// ---------------------------------------------------------------------------
// Kernel 2: one workgroup per bag.
//   Phase 1: per-sentence logits -> global workspace
//   Phase 2: bag max + sum(exp)
//   Phase 3: bag_repre[b,d] = sum_i softmax_w[i] * x[i,d]
// ---------------------------------------------------------------------------
__global__ __launch_bounds__(BLK) void bag_kernel(
    const float* __restrict__ x,
    const float* __restrict__ combined,
    const int* __restrict__ query,
    const int* __restrict__ scope,
    float* __restrict__ ws_logit,
    float* __restrict__ bag_repre) {

    __shared__ float s_red[NWAVES];
    __shared__ float s_bcast[2];
    __shared__ float s_w[WCHUNK];

    const int b = blockIdx.x;
    const int start = scope[b];
    const int end = scope[b + 1];
    const int n = end - start;
    const int tid = threadIdx.x;
    const int lane = tid & 31;
    const int wave = tid >> 5;

    // ---- Phase 1: logits (one wave per sentence) ----
    for (int s = wave; s < n; s += NWAVES) {
        const int i = start + s;
        const int q = query[i];
        const float* xr = x + (size_t)i * REL_DIM;
        const float* cr = combined + (size_t)q * REL_DIM;
        float acc = 0.f;
        for (int d = lane; d < REL_DIM; d += 32) acc += xr[d] * cr[d];
        #pragma unroll
        for (int o = 16; o > 0; o >>= 1) acc += __shfl_xor(acc, o, 32);
        if (lane == 0) ws_logit[i] = acc;
    }
    __threadfence();
    __syncthreads();

    // ---- Phase 2a: max ----
    float m = -INFINITY;
    for (int s = tid; s < n; s += BLK) m = fmaxf(m, ws_logit[start + s]);
    #pragma unroll
    for (int o = 16; o > 0; o >>= 1) m = fmaxf(m, __shfl_xor(m, o, 32));
    if (lane == 0) s_red[wave] = m;
    __syncthreads();
    if (tid == 0) {
        float mm = s_red[0];
        for (int w = 1; w < NWAVES; w++) mm = fmaxf(mm, s_red[w]);
        s_bcast[0] = mm;
    }
    __syncthreads();
    const float M = s_bcast[0];

    // ---- Phase 2b: sum(exp) ----
    float sum = 0.f;
    for (int s = tid; s < n; s += BLK) sum += __expf(ws_logit[start + s] - M);
    #pragma unroll
    for (int o = 16; o > 0; o >>= 1) sum += __shfl_xor(sum, o, 32);
    __syncthreads();
    if (lane == 0) s_red[wave] = sum;
    __syncthreads();
    if (tid == 0) {
        float ss = 0.f;
        for (int w = 0; w < NWAVES; w++) ss += s_red[w];
        s_bcast[1] = ss;
    }
    __syncthreads();
    const float inv = 1.f / s_bcast[1];

    // ---- Phase 3: weighted feature sum (each thread owns <=3 feature cols) ----
    float acc0 = 0.f, acc1 = 0.f, acc2 = 0.f;
    const int d0 = tid, d1 = tid + BLK, d2 = tid + 2 * BLK;
    for (int s0 = 0; s0 < n; s0 += WCHUNK) {
        const int cn = min(WCHUNK, n - s0);
        __syncthreads();
        for (int s = tid; s < cn; s += BLK)
            s_w[s] = __expf(ws_logit[start + s0 + s] - M) * inv;
        __syncthreads();
        for (int s = 0; s < cn; s++) {
            const float* xr = x + (size_t)(start + s0 + s) * REL_DIM;
            const float w = s_w[s];
            acc0 += w * xr[d0];
            acc1 += w * xr[d1];
            if (d2 < REL_DIM) acc2 += w * xr[d2];
        }
    }
    float* br = bag_repre + (size_t)b * REL_DIM;
    br[d0] = acc0;
    br[d1] = acc1;
    if (d2 < REL_DIM) br[d2] = acc2;
}

// ---------------------------------------------------------------------------
// Kernel 3: out[b,c] = bag_repre[b,:] . relation_weight[c,:] + bias[c]
// WMMA f32 16x16x4: each wave owns one 16x16 output tile.
//   A lane layout: row = lane%16, K pair = (lane/16)*2      (2 VGPRs = float2)
//   B lane layout: col = lane%16, K pair = (lane/16)*2      (2 VGPRs = float2)
//   C/D layout   : VGPR v -> row v + 8*(lane/16), col = lane%16
// ---------------------------------------------------------------------------
__global__ __launch_bounds__(128) void gemm_wmma(
    const float* __restrict__ bag,
    const float* __restrict__ rw,
    const float* __restrict__ bias,
    float* __restrict__ out, int B) {

    const int mtile = blockIdx.x;
    const int wave = threadIdx.x >> 5;   // 0..3 -> column tile
    const int lane = threadIdx.x & 31;
    const int row0 = mtile * 16;
    const int col0 = wave * 16;

    const int sub = lane & 15;
    const int khalf = (lane >> 4) * 2;   // 0 or 2

    const float* arow = bag + (size_t)(row0 + sub) * REL_DIM;
    const int colIdx = col0 + sub;
    const bool colValid = colIdx < NUM_CLASSES;
    const float* brow = rw + (size_t)(colValid ? colIdx : 0) * REL_DIM;

    v8f acc = {};
    // main K loop: k = 0..684 step 4 (covers 688 of 690)
    for (int k = 0; k < REL_DIM - 2; k += 4) {
        v2f a = *(const v2f*)(arow + k + khalf);
        v2f bf = *(const v2f*)(brow + k + khalf);
        acc = __builtin_amdgcn_wmma_f32_16x16x4_f32(
            false, a, false, bf, (short)0, acc, false, false);
    }
    // remainder: k = 688, only khalf==0 lanes carry data (K=688,689)
    {
        v2f a = {0.f, 0.f};
        v2f bf = {0.f, 0.f};
        if (khalf == 0) {
            a = *(const v2f*)(arow + (REL_DIM - 2));
            bf = *(const v2f*)(brow + (REL_DIM - 2));
        }
        acc = __builtin_amdgcn_wmma_f32_16x16x4_f32(
            false, a, false, bf, (short)0, acc, false, false);
    }

    if (colValid) {
        const float bv = bias[colIdx];
        const int rbase = row0 + ((lane >> 4) * 8);
        #pragma unroll
        for (int v = 0; v < 8; v++) {
            const int r = rbase + v;
            if (r < B) out[(size_t)r * NUM_CLASSES + colIdx] = acc[v] + bv;
        }
    }
}

// ---------------------------------------------------------------------------
extern "C" void kernel_launch(void* const* d_in, const int* in_sizes, int n_in,
                              void* d_out, int out_size, void* d_ws, size_t ws_size,
                              hipStream_t stream) {
    const float* x    = (const float*)d_in[0];
    const float* relw = (const float*)d_in[1];
    const float* attw = (const float*)d_in[2];
    const float* bias = (const float*)d_in[3];
    const int* query  = (const int*)d_in[4];
    const int* scope  = (const int*)d_in[5];

    const int N = in_sizes[4];          // 131072 sentences
    const int B = in_sizes[5] - 1;      // 8192 bags
    const int C = in_sizes[3];          // 53 classes (== NUM_CLASSES)

    // workspace layout (floats): [bag_repre B*D | ws_logit N | combined C*D]
    float* bag_repre = (float*)d_ws;
    float* ws_logit  = bag_repre + (size_t)B * REL_DIM;
    float* combined  = ws_logit + (size_t)N;

    // 1) combined = att * rel
    {
        const int n = C * REL_DIM;
        combine_kernel<<<(n + 255) / 256, 256, 0, stream>>>(attw, relw, combined, n);
    }
    // 2) per-bag softmax-weighted representation
    bag_kernel<<<B, BLK, 0, stream>>>(x, combined, query, scope, ws_logit, bag_repre);
    // 3) WMMA GEMM: [B x D] x [D x C] + bias
    {
        const int mtiles = (B + 15) / 16;
        gemm_wmma<<<mtiles, 128, 0, stream>>>(bag_repre, relw, bias, (float*)d_out, B);
    }
}